// ConvBKI_38774964748738
// MI455X (gfx1250) — compile-verified
//
#include <hip/hip_runtime.h>
#include <hip/hip_bf16.h>

// ---------------- problem constants ----------------
#define GXD 256
#define GYD 256
#define GZD 32
#define NC  21
#define COLS (GZD*NC)                    // 672 floats per (x,y) column
#define NVOX ((size_t)GXD*GYD*COLS)      // 44,040,192 floats
#define NPTS 500000
#define PT_STRIDE 24                     // xyz + 21 labels
#define VOXSZ 0.15625f                   // identical for x,y,z: 40/256 = 5/32
#define MINB_X (-20.0f)
#define MINB_Y (-20.0f)
#define MINB_Z (-2.5f)
#define MAXB_X (20.0f)
#define MAXB_Y (20.0f)
#define MAXB_Z (2.5f)
#define PI_F 3.14159265358979323846f

typedef float v2f __attribute__((ext_vector_type(2)));
typedef float v8f __attribute__((ext_vector_type(8)));

#if defined(__gfx1250__) && __has_builtin(__builtin_amdgcn_global_load_async_to_lds_b32)
#define HAS_ASYNC_LDS 1
typedef __attribute__((address_space(1))) int* as1_i32p;
typedef __attribute__((address_space(3))) int* as3_i32p;
#else
#define HAS_ASYNC_LDS 0
#endif

// ---------------- kernel 0: zero the scatter grid ----------------
__global__ void zero_grid(float4* __restrict__ p, int n4) {
  int i = blockIdx.x * blockDim.x + threadIdx.x;
  int stride = gridDim.x * blockDim.x;
  for (; i < n4; i += stride) p[i] = make_float4(0.f, 0.f, 0.f, 0.f);
}

// ---------------- kernel 1: sparse-kernel weights (125) ----------------
__global__ void kv_eval(const float* __restrict__ ell_p, float* __restrict__ kv) {
  int t = threadIdx.x;
  if (t >= 125) return;
  float ell = ell_p[0];
  int i = t / 25, j = (t / 5) % 5, k = t % 5;
  float di = fabsf((float)i - 2.0f);
  float dj = fabsf((float)j - 2.0f);
  float dk = fabsf((float)k - 2.0f);
  float d = VOXSZ * sqrtf(di * di + dj * dj + dk * dk);
  float r = d / ell;
  float tt = 2.0f * PI_F * r;
  float val = (2.0f + cosf(tt)) * (1.0f - r) * (1.0f / 3.0f) + sinf(tt) * (1.0f / (2.0f * PI_F));
  val = (d >= ell) ? 0.0f : val;
  val = fminf(fmaxf(val, 0.0f), 1.0f);
  kv[t] = val;
}

// ---------------- kernel 2: point scatter-add ----------------
__global__ void scatter_points(const float* __restrict__ pc, float* __restrict__ upd, int n) {
  int p = blockIdx.x * blockDim.x + threadIdx.x;
  if (p >= n) return;
  const float* q = pc + (size_t)p * PT_STRIDE;
  float x = q[0], y = q[1], z = q[2];
  bool valid = (x >= MINB_X) && (x < MAXB_X) &&
               (y >= MINB_Y) && (y < MAXB_Y) &&
               (z >= MINB_Z) && (z < MAXB_Z);
  if (!valid) return;
  int gx = (int)floorf((x - MINB_X) / VOXSZ);
  int gy = (int)floorf((y - MINB_Y) / VOXSZ);
  int gz = (int)floorf((z - MINB_Z) / VOXSZ);
  gx = min(max(gx, 0), GXD - 1);
  gy = min(max(gy, 0), GYD - 1);
  gz = min(max(gz, 0), GZD - 1);
  size_t base = ((size_t)(gx * GYD + gy)) * COLS + (size_t)gz * NC;
  #pragma unroll
  for (int c = 0; c < NC; ++c) atomicAdd(&upd[base + c], q[3 + c]);
}

// ---------------- kernel 3: depthwise 5x5x5 conv via f32 WMMA ----------------
// Block: 4x4 xy output patch, all 32 z, all 21 classes (classes phased in 3 groups of 7).
// WMMA f32 16x16x4:  M = 16 output-z, N = 16 patch columns, K = input-z chunk of 4.
// z-conv = banded 32x32 Toeplitz; 5 K-chunks cover band +-2 of each 16-z tile.
#define CSTRIDE 225   // padded column stride in LDS (224 data floats + 1 pad -> conflict-free)
__global__ __launch_bounds__(256)
void conv_bki(const float* __restrict__ cmap, const float* __restrict__ upd,
              const float* __restrict__ kv, float* __restrict__ out) {
  __shared__ float s_in[64 * CSTRIDE];   // 8x8 columns x 32 z x 7 classes (padded) = 14,400 floats
  __shared__ float s_kv[128];

  const int tid  = threadIdx.x;
  const int lane = tid & 31;
  const int wave = tid >> 5;

  if (tid < 125) s_kv[tid] = kv[tid];

  const int x0 = blockIdx.x * 4;
  const int y0 = blockIdx.y * 4;

  const int ztile   = wave & 1;        // z 0..15 or 16..31
  const int cmod    = wave >> 1;       // class residue 0..3 within group of 7
  const int n       = lane & 15;       // N index: patch column
  const int px      = n >> 2;
  const int py      = n & 3;
  const int halfsel = lane >> 4;       // upper half of wave
  const int kk      = halfsel << 1;    // K offset {0,2} per ISA A/B layouts
  const int kcbase  = ztile * 3;       // k-chunks {0..4} or {3..7}: zi in [0,19] / [12,31]

  for (int cg = 0; cg < 3; ++cg) {
    __syncthreads();                   // s_in free for reuse (prev writeback done)
    // ---- halo-tile load: 8x8 columns x 32 z x 7 classes -> LDS ----
    for (int idx = tid; idx < 64 * 224; idx += 256) {
      int col = idx / 224;
      int rem = idx - col * 224;            // z*7 + j
      int z   = rem / 7;
      int j   = rem - z * 7;
      int gx  = x0 - 2 + (col >> 3);
      int gy  = y0 - 2 + (col & 7);
      int li  = col * CSTRIDE + rem;
#if HAS_ASYNC_LDS
      if ((unsigned)gx < GXD && (unsigned)gy < GYD) {
        const float* g = &upd[((size_t)((gx << 8) + gy)) * COLS + z * NC + cg * 7 + j];
        __builtin_amdgcn_global_load_async_to_lds_b32(
            (as1_i32p)(size_t)g,                 // AS1 int*, const shed via integer
            (as3_i32p)&s_in[li],                 // generic -> local addrspacecast
            0, 0);
      } else {
        s_in[li] = 0.0f;
      }
#else
      float v = 0.0f;
      if ((unsigned)gx < GXD && (unsigned)gy < GYD)
        v = upd[((size_t)((gx << 8) + gy)) * COLS + z * NC + cg * 7 + j];
      s_in[li] = v;
#endif
    }
#if HAS_ASYNC_LDS
#if __has_builtin(__builtin_amdgcn_s_wait_asynccnt)
    __builtin_amdgcn_s_wait_asynccnt(0);
#else
    asm volatile("s_wait_asynccnt 0x0" ::: "memory");
#endif
#endif
    __syncthreads();

    const int j0 = cmod;                 // this wave's classes within the group
    const int j1 = cmod + 4;
    const bool two = (j1 < 7);           // wave-uniform

    v8f acc0 = {0.f, 0.f, 0.f, 0.f, 0.f, 0.f, 0.f, 0.f};
    v8f acc1 = {0.f, 0.f, 0.f, 0.f, 0.f, 0.f, 0.f, 0.f};

    for (int dx = 0; dx < 5; ++dx) {
      for (int dy = 0; dy < 5; ++dy) {
        const float* wz = &s_kv[(dx * 5 + dy) * 5];
        const float w0 = wz[0], w1 = wz[1], w2 = wz[2], w3 = wz[3], w4 = wz[4];
        // Build the 5 banded-Toeplitz A blocks for this (dx,dy) pass.
        v2f A[5];
        #pragma unroll
        for (int i = 0; i < 5; ++i) {
          #pragma unroll
          for (int v = 0; v < 2; ++v) {
            int zi = (kcbase + i) * 4 + kk + v;     // input z (always in [0,31])
            int zo = ztile * 16 + n;                // output z (M = lane&15)
            int dd = zi - zo + 2;                   // tap index
            float a = (dd == 0) ? w0 : (dd == 1) ? w1 : (dd == 2) ? w2
                    : (dd == 3) ? w3 : (dd == 4) ? w4 : 0.0f;
            A[i][v] = a;
          }
        }
        const int colB = ((px + dx) << 3) + (py + dy);
        const float* bbase = &s_in[colB * CSTRIDE];

        { // class j0
          const float* bb = bbase + j0;
          #pragma unroll
          for (int i = 0; i < 5; ++i) {
            int zi0 = (kcbase + i) * 4 + kk;
            v2f B;
            B[0] = bb[zi0 * 7];
            B[1] = bb[(zi0 + 1) * 7];
            acc0 = __builtin_amdgcn_wmma_f32_16x16x4_f32(false, A[i], false, B,
                                                         (short)0, acc0, false, false);
          }
        }
        if (two) { // class j1 (wave-uniform branch: EXEC stays all-1s)
          const float* bb = bbase + j1;
          #pragma unroll
          for (int i = 0; i < 5; ++i) {
            int zi0 = (kcbase + i) * 4 + kk;
            v2f B;
            B[0] = bb[zi0 * 7];
            B[1] = bb[(zi0 + 1) * 7];
            acc1 = __builtin_amdgcn_wmma_f32_16x16x4_f32(false, A[i], false, B,
                                                         (short)0, acc1, false, false);
          }
        }
      }
    }

    __syncthreads();   // all waves finished reading s_in for this class group

    // ---- stage accumulators into LDS: [col 0..15][z 0..31][j 0..6] (padded 225) ----
    {
      #pragma unroll
      for (int r = 0; r < 8; ++r) {
        int z = ztile * 16 + r + (halfsel << 3);    // C/D layout: M = r + 8*(lane>=16)
        s_in[n * CSTRIDE + z * 7 + j0] = acc0[r];
      }
      if (two) {
        #pragma unroll
        for (int r = 0; r < 8; ++r) {
          int z = ztile * 16 + r + (halfsel << 3);
          s_in[n * CSTRIDE + z * 7 + j1] = acc1[r];
        }
      }
    }
    __syncthreads();

    // ---- coalesced writeback (7-float runs): out = current_map + conv ----
    for (int idx = tid; idx < 16 * 224; idx += 256) {
      int col = idx / 224;                  // patch column 0..15
      int rem = idx - col * 224;            // z*7 + j
      int z   = rem / 7;
      int j   = rem - z * 7;
      size_t g = ((size_t)(((x0 + (col >> 2)) << 8) + (y0 + (col & 3)))) * COLS
               + (size_t)z * NC + cg * 7 + j;
      float cm = __builtin_nontemporal_load(&cmap[g]);
      __builtin_nontemporal_store(cm + s_in[col * CSTRIDE + rem], &out[g]);
    }
  }
}

// ---------------- host launcher ----------------
extern "C" void kernel_launch(void* const* d_in, const int* in_sizes, int n_in,
                              void* d_out, int out_size, void* d_ws, size_t ws_size,
                              hipStream_t stream) {
  const float* cmap = (const float*)d_in[0];   // 256*256*32*21
  const float* pc   = (const float*)d_in[1];   // 500000*24
  const float* ell  = (const float*)d_in[2];   // scalar
  float* out = (float*)d_out;
  float* upd = (float*)d_ws;                   // 44,040,192 floats scatter grid
  float* kvb = upd + NVOX;                     // 125 kernel weights

  zero_grid<<<2048, 256, 0, stream>>>((float4*)upd, (int)(NVOX / 4));
  kv_eval<<<1, 128, 0, stream>>>(ell, kvb);
  scatter_points<<<(NPTS + 255) / 256, 256, 0, stream>>>(pc, upd, NPTS);
  dim3 grid(GXD / 4, GYD / 4);
  conv_bki<<<grid, 256, 0, stream>>>(cmap, upd, kvb, out);
}